// TorchModel_59219009077447
// MI455X (gfx1250) — compile-verified
//
#include <hip/hip_runtime.h>
#include <hip/hip_bf16.h>

typedef __attribute__((ext_vector_type(2))) float v2f;
typedef __attribute__((ext_vector_type(8))) float v8f;

#ifndef __has_builtin
#define __has_builtin(x) 0
#endif
#if __has_builtin(__builtin_amdgcn_tanhf)
#define TANH_F32(x) __builtin_amdgcn_tanhf(x)
#else
#define TANH_F32(x) tanhf(x)
#endif

namespace {
constexpr int kDim  = 20;
constexpr int kHid  = 20;
constexpr int kNcls = 5;
constexpr int kB    = 16384;
constexpr int kS    = 256;
}

// D = A(16x4 f32) * B(4x16 f32) + C(16x16 f32)
__device__ __forceinline__ v8f wmma4(v2f a, v2f b, v8f c) {
    return __builtin_amdgcn_wmma_f32_16x16x4_f32(
        /*neg_a=*/false, a, /*neg_b=*/false, b,
        /*c_mod=*/(short)0, c, /*reuse_a=*/false, /*reuse_b=*/false);
}

// Load one 16-row A-operand tile from a row-major [nrows x 20] weight matrix.
// 'wrow' is the (already permuted) source row for this lane; 'valid' masks
// padding rows to zero without touching EXEC.
__device__ __forceinline__ void load_A_rows(const float* __restrict__ W, int wrow, bool valid,
                                            int khalf, v2f A[5]) {
    int rc = valid ? wrow : 0;  // clamp to a real row, mask after load
#pragma unroll
    for (int s = 0; s < 5; ++s) {
        int k0 = 4 * s + khalf;
        float2 w = *(const float2*)(W + rc * kDim + k0);
        A[s].x = valid ? w.x : 0.0f;
        A[s].y = valid ? w.y : 0.0f;
    }
}

// Gather one timestep of embeddings straight into B-operand layout:
// lane L (half = L/16) supplies (K = 4s+2*half, 4s+2*half+1) for batch col L%16.
__device__ __forceinline__ void load_e(const int* __restrict__ x, const float* __restrict__ emb,
                                       int bglob, int t, int khalf, float Be[10]) {
    int idx = x[bglob * kS + t];
    const float* row = emb + (long)idx * kDim;
    bool pad = (idx == 0);                 // padding_idx=0 -> zero row
#pragma unroll
    for (int s = 0; s < 5; ++s) {
        float2 v = *(const float2*)(row + 4 * s + khalf);
        Be[2 * s]     = pad ? 0.0f : v.x;
        Be[2 * s + 1] = pad ? 0.0f : v.y;
    }
}

__global__ __launch_bounds__(256) void rnn_fused_kernel(
    const int*   __restrict__ x,      // [B, S]
    const float* __restrict__ emb,    // [VOCAB, 20]
    const float* __restrict__ W_ih,   // [20, 20]
    const float* __restrict__ W_hh,   // [20, 20]
    const float* __restrict__ W_cls,  // [5, 20]
    const float* __restrict__ b_cls,  // [5]
    float*       __restrict__ out)    // [B, 5]
{
    const int lane   = threadIdx.x & 31;
    const int wave   = threadIdx.x >> 5;
    const int batch0 = (blockIdx.x * 8 + wave) * 16;   // 16 batch rows per wave
    const int  row   = lane & 15;
    const int  khalf = (lane >> 4) * 2;
    const int  bglob = batch0 + row;

    // --- Row permutation that makes D (after tanh) directly reusable as the
    // --- next step's B operand (see analysis):
    //   tile0: D row m holds h[k], k = 4*((m&7)>>1) + (m&1) + 2*(m>>3)
    //   tile1: A rows {0,1,8,9} <- W rows {16,17,18,19}; rest zero
    const int  k0row   = 4 * ((row & 7) >> 1) + (row & 1) + 2 * (row >> 3);
    const int  k1row   = 16 + (row & 1) + 2 * (row >> 3);
    const bool t1valid = ((row & 6) == 0);

    // Loop-invariant A operands (weights), preloaded once.
    v2f Aih0[5], Aih1[5], Ahh0[5], Ahh1[5], Acls[5];
    load_A_rows(W_ih, k0row, true,      khalf, Aih0);
    load_A_rows(W_ih, k1row, t1valid,   khalf, Aih1);
    load_A_rows(W_hh, k0row, true,      khalf, Ahh0);
    load_A_rows(W_hh, k1row, t1valid,   khalf, Ahh1);
    load_A_rows(W_cls, row,  row < kNcls, khalf, Acls);   // classifier: no permutation

    // State h^T, stored permuted so that:
    //   slice s (K=4s..4s+3) B operand = {h0[2s], h0[2s+1]},  slice 4 = {h1[0], h1[1]}
    v8f h0 = {};
    float h1a = 0.0f, h1b = 0.0f;

    float BeC[10];
    load_e(x, emb, bglob, 0, khalf, BeC);

    for (int t = 0; t < kS; ++t) {
        // Prefetch next timestep's embeddings (independent of the h-chain).
        float BeN[10];
        int tn = (t + 1 < kS) ? (t + 1) : t;
        load_e(x, emb, bglob, tn, khalf, BeN);

        // acc = W_ih*e^T first (h-independent head of the chain) ...
        v8f acc0 = {};
        v8f acc1 = {};
#pragma unroll
        for (int s = 0; s < 5; ++s) {
            v2f be; be.x = BeC[2 * s]; be.y = BeC[2 * s + 1];
            acc0 = wmma4(Aih0[s], be, acc0);
            acc1 = wmma4(Aih1[s], be, acc1);
        }
        // ... then W_hh*h^T (h-dependent tail), B operand read straight from state.
#pragma unroll
        for (int s = 0; s < 5; ++s) {
            v2f bh;
            if (s < 4) { bh.x = h0[2 * s]; bh.y = h0[2 * s + 1]; }
            else       { bh.x = h1a;       bh.y = h1b;           }
            acc0 = wmma4(Ahh0[s], bh, acc0);
            acc1 = wmma4(Ahh1[s], bh, acc1);
        }

        // h = tanh(acc): 8 state values in tile0, 2 in tile1 (rest are zero rows).
        v8f nh0;
#pragma unroll
        for (int j = 0; j < 8; ++j) nh0[j] = TANH_F32(acc0[j]);
        h0  = nh0;
        h1a = TANH_F32(acc1[0]);
        h1b = TANH_F32(acc1[1]);

#pragma unroll
        for (int i = 0; i < 10; ++i) BeC[i] = BeN[i];
    }

    // Classifier: y^T[c][b] = W_cls * h^T  (values in B layout are unpermuted h[k]).
    v8f yc = {};
#pragma unroll
    for (int s = 0; s < 5; ++s) {
        v2f bh;
        if (s < 4) { bh.x = h0[2 * s]; bh.y = h0[2 * s + 1]; }
        else       { bh.x = h1a;       bh.y = h1b;           }
        yc = wmma4(Acls[s], bh, yc);
    }

    if (lane < 16) {
#pragma unroll
        for (int c = 0; c < kNcls; ++c) {
            out[bglob * kNcls + c] = yc[c] + b_cls[c];
        }
    }
}

extern "C" void kernel_launch(void* const* d_in, const int* in_sizes, int n_in,
                              void* d_out, int out_size, void* d_ws, size_t ws_size,
                              hipStream_t stream) {
    const int*   x     = (const int*)d_in[0];
    const float* emb   = (const float*)d_in[1];
    const float* W_ih  = (const float*)d_in[2];
    const float* W_hh  = (const float*)d_in[3];
    const float* W_cls = (const float*)d_in[4];
    const float* b_cls = (const float*)d_in[5];
    float* out = (float*)d_out;

    dim3 block(256);                    // 8 waves x 16 batch rows = 128 rows/block
    dim3 grid(kB / 128);                // 128 blocks, 1024 waves total
    hipLaunchKernelGGL(rnn_fused_kernel, grid, block, 0, stream,
                       x, emb, W_ih, W_hh, W_cls, b_cls, out);
}